// RadialBasisFunctionPotential_86062554677256
// MI455X (gfx1250) — compile-verified
//
#include <hip/hip_runtime.h>

typedef float v2f __attribute__((ext_vector_type(2)));
typedef float v8f __attribute__((ext_vector_type(8)));

#define ANCHOR_CHUNK 4096            // anchors staged per LDS pass (64 KB as float4)
#define BLOCK_THREADS 256            // 8 wave32s -> 128 rows of N per block

// RBF potential: out[n] = sum_m coeff[m] * exp(-0.5*||x_n - a_m||^2 / w^2)
//
// One V_WMMA_F32_16X16X4_F32 per 16x16 (n,m) tile produces cexp*d2 directly:
//   A[i,0..2] = -2*cexp * x_i ,  A[i,3] = 1
//   B[0..2,j] =  a_j            B[3,j]  = cexp * |a_j|^2
//   C[i,j]    =  cexp * |x_i|^2                     (loop-invariant!)
//   => D[i,j] = cexp * ||x_i - a_j||^2 ,  k = exp2(D)
// with cexp = -0.5 * log2(e) / w^2.
__global__ __launch_bounds__(BLOCK_THREADS)
void rbf_wmma_f32_kernel(const float* __restrict__ x,      // [N,3]
                         const float* __restrict__ anc,    // [M,3]
                         const float* __restrict__ coeff,  // [M]
                         const float* __restrict__ widthp, // [1]
                         float* __restrict__ out,          // [N]
                         int N, int M)
{
    __shared__ float4 sm[ANCHOR_CHUNK];   // {ax, ay, az, coeff}

    const int tid  = threadIdx.x;
    const int lane = tid & 31;
    const int wave = tid >> 5;
    const int half = lane >> 4;           // 0: lanes 0-15, 1: lanes 16-31
    const int l16  = lane & 15;

    const int n0 = (blockIdx.x * (BLOCK_THREADS / 32) + wave) * 16;

    const float w    = widthp[0];
    const float cexp = (-0.5f / (w * w)) * 1.44269504088896340736f; // -0.5*log2(e)/w^2
    const float s    = -2.0f * cexp;

    // ---- A-matrix (2 VGPRs):
    // lanes 0-15:  row=l16, (K0,K1) = (s*x, s*y)
    // lanes 16-31: row=l16, (K2,K3) = (s*z, 1)
    v2f A;
    {
        int r = n0 + l16; if (r >= N) r = N - 1;
        float cx = x[3 * r + 0];
        float cy = x[3 * r + 1];
        float cz = x[3 * r + 2];
        A.x = s * (half ? cz : cx);
        A.y = half ? 1.0f : (s * cy);
    }

    // ---- C-matrix: cexp * |x_row|^2 for the 8 D-rows this half covers (constant)
    v8f C;
#pragma unroll
    for (int v = 0; v < 8; ++v) {
        int r = n0 + half * 8 + v; if (r >= N) r = N - 1;
        float cx = x[3 * r + 0];
        float cy = x[3 * r + 1];
        float cz = x[3 * r + 2];
        C[v] = cexp * fmaf(cx, cx, fmaf(cy, cy, cz * cz));
    }

    float acc[8];
#pragma unroll
    for (int v = 0; v < 8; ++v) acc[v] = 0.0f;

    for (int mbase = 0; mbase < M; mbase += ANCHOR_CHUNK) {
        // ---- cooperative LDS fill (coalesced; zero-pad OOB anchors via coeff=0)
        for (int i = tid; i < ANCHOR_CHUNK; i += BLOCK_THREADS) {
            int m = mbase + i;
            float4 t;
            if (m < M) {
                t.x = anc[3 * m + 0];
                t.y = anc[3 * m + 1];
                t.z = anc[3 * m + 2];
                t.w = coeff[m];
            } else {
                t.x = 0.0f; t.y = 0.0f; t.z = 0.0f; t.w = 0.0f;
            }
            sm[i] = t;
        }
        __syncthreads();

        // ---- sweep 16-anchor tiles from LDS
        for (int mt = 0; mt < ANCHOR_CHUNK; mt += 16) {
            float4 an = sm[mt + l16];     // this lane's anchor (m = mt + l16)

            float a2c = cexp * fmaf(an.x, an.x, fmaf(an.y, an.y, an.z * an.z));

            // B-matrix: lanes 0-15 (K0,K1)=(ax,ay); lanes 16-31 (K2,K3)=(az, cexp*a2)
            v2f B;
            B.x = half ? an.z : an.x;
            B.y = half ? a2c  : an.y;

            // D = cexp * ||x - a||^2   (single matrix op, no per-element prep)
            v8f D = __builtin_amdgcn_wmma_f32_16x16x4_f32(
                false, A, false, B, (short)0, C, false, false);

#pragma unroll
            for (int v = 0; v < 8; ++v) {
                float k = __builtin_amdgcn_exp2f(D[v]);   // v_exp_f32
                acc[v]  = fmaf(k, an.w, acc[v]);
            }
        }
        __syncthreads();
    }

    // ---- reduce across the 16 lanes of each half (anchors residue-split by lane)
#pragma unroll
    for (int v = 0; v < 8; ++v) {
        float sum = acc[v];
        sum += __shfl_xor(sum, 1, 32);
        sum += __shfl_xor(sum, 2, 32);
        sum += __shfl_xor(sum, 4, 32);
        sum += __shfl_xor(sum, 8, 32);
        acc[v] = sum;
    }

    if (l16 == 0) {
        int rbase = n0 + half * 8;
#pragma unroll
        for (int v = 0; v < 8; ++v) {
            int r = rbase + v;
            if (r < N) out[r] = acc[v];
        }
    }
}

extern "C" void kernel_launch(void* const* d_in, const int* in_sizes, int n_in,
                              void* d_out, int out_size, void* d_ws, size_t ws_size,
                              hipStream_t stream) {
    (void)n_in; (void)d_ws; (void)ws_size; (void)out_size;

    const float* x      = (const float*)d_in[0];  // [N,3]
    const float* anc    = (const float*)d_in[1];  // [M,3]
    const float* coeff  = (const float*)d_in[2];  // [M]
    const float* widthp = (const float*)d_in[3];  // [1]
    float* out          = (float*)d_out;          // [N]

    const int N = in_sizes[0] / 3;
    const int M = in_sizes[1] / 3;

    const int rows_per_block = (BLOCK_THREADS / 32) * 16;   // 128
    const int grid = (N + rows_per_block - 1) / rows_per_block;

    rbf_wmma_f32_kernel<<<grid, BLOCK_THREADS, 0, stream>>>(x, anc, coeff, widthp, out, N, M);
}